// YoloLoss_57406532878592
// MI455X (gfx1250) — compile-verified
//
#include <hip/hip_runtime.h>

// YOLO loss: S=7, B=2, C=20; pred cell = 30 f32, target cell = 25 f32.
// Bandwidth-bound streaming reduction; WMMA f32 16x16x4 used for the
// wave32 cross-lane reduction (B = ones => D rows hold A row-sums).

typedef float v2f __attribute__((ext_vector_type(2)));
typedef float v8f __attribute__((ext_vector_type(8)));

__device__ __forceinline__ float sqf(float x) { return x * x; }

__device__ __forceinline__ float cell_loss(const float* __restrict__ p,
                                           const float* __restrict__ t) {
    // predictions: base + cell*120B -> 8-byte aligned, use 64-bit NT loads
    float pr[30];
    const v2f* pv = (const v2f*)p;
#pragma unroll
    for (int i = 0; i < 15; ++i) {
        v2f v = __builtin_nontemporal_load(pv + i);
        pr[2 * i]     = v.x;
        pr[2 * i + 1] = v.y;
    }
    // target: base + cell*100B -> only 4-byte aligned, scalar NT loads
    float tg[25];
#pragma unroll
    for (int i = 0; i < 25; ++i) tg[i] = __builtin_nontemporal_load(t + i);

    const float exists = tg[20];
    const float tx = tg[21], ty = tg[22], tw = tg[23], th = tg[24];
    const float tx1 = tx - 0.5f * tw, ty1 = ty - 0.5f * th;
    const float tx2 = tx + 0.5f * tw, ty2 = ty + 0.5f * th;
    const float areaT = fabsf((tx2 - tx1) * (ty2 - ty1));

    float iou[2], cf[2], bx[2], by[2], bw[2], bh[2];
#pragma unroll
    for (int b = 0; b < 2; ++b) {
        const int o = 20 + 5 * b;
        cf[b] = pr[o];
        bx[b] = pr[o + 1]; by[b] = pr[o + 2];
        bw[b] = pr[o + 3]; bh[b] = pr[o + 4];
        const float x1 = bx[b] - 0.5f * bw[b], y1 = by[b] - 0.5f * bh[b];
        const float x2 = bx[b] + 0.5f * bw[b], y2 = by[b] + 0.5f * bh[b];
        const float ix1 = fmaxf(x1, tx1), iy1 = fmaxf(y1, ty1);
        const float ix2 = fminf(x2, tx2), iy2 = fminf(y2, ty2);
        const float inter = fmaxf(ix2 - ix1, 0.0f) * fmaxf(iy2 - iy1, 0.0f);
        const float a1 = fabsf((x2 - x1) * (y2 - y1));
        iou[b] = inter / (a1 + areaT - inter + 1e-6f);
    }
    // jnp.argmax picks first max on tie -> best=1 only if strictly greater
    const int best = (iou[1] > iou[0]) ? 1 : 0;
    const float pcf = cf[best], pbx = bx[best], pby = by[best];
    const float pbw = bw[best], pbh = bh[best];

    const float ew = exists * pbw, eh = exists * pbh;
    const float sw = (ew > 0.0f) ? 1.0f : ((ew < 0.0f) ? -1.0f : 0.0f);
    const float sh = (eh > 0.0f) ? 1.0f : ((eh < 0.0f) ? -1.0f : 0.0f);
    const float bpw = sw * sqrtf(fabsf(ew) + 1.0e6f);  // EPS_BUG
    const float bph = sh * sqrtf(fabsf(eh) + 1.0e6f);

    float box_loss = sqf(exists * pbx - exists * tx)
                   + sqf(exists * pby - exists * ty)
                   + sqf(bpw - sqrtf(exists * tw))
                   + sqf(bph - sqrtf(exists * th));

    const float obj = sqf(exists * pcf - exists * exists);  // target conf == exists
    const float om = 1.0f - exists;
    const float noobj = sqf(om * cf[0] - om * exists) + sqf(om * cf[1] - om * exists);

    float cls = 0.0f;
#pragma unroll
    for (int c = 0; c < 20; ++c) cls += sqf(exists * pr[c] - exists * tg[c]);

    return 5.0f * box_loss + obj + 0.5f * noobj + cls;
}

__global__ void __launch_bounds__(256)
yolo_loss_main(const float* __restrict__ pred, const float* __restrict__ targ,
               float* __restrict__ block_sums, int n_cells) {
    float acc = 0.0f;
    const int stride = gridDim.x * blockDim.x;
    for (int cell = blockIdx.x * blockDim.x + threadIdx.x; cell < n_cells;
         cell += stride) {
        acc += cell_loss(pred + (size_t)cell * 30, targ + (size_t)cell * 25);
    }

    // ---- wave32 reduction through the matrix pipe ----
    // A(16x4): lane L<16 holds {acc_L, 0} at K0/K1; lane L>=16 holds
    // {acc_L, 0} at K2/K3 of row L-16.  B = ones(4x16).
    // D[m][n] = rowsum(m) = acc[m] + acc[m+16], replicated over n.
    v2f A;  A.x = acc;  A.y = 0.0f;
    v2f Bm; Bm.x = 1.0f; Bm.y = 1.0f;
    v8f Cz = {};
    v8f D = __builtin_amdgcn_wmma_f32_16x16x4_f32(
        /*neg_a=*/false, A, /*neg_b=*/false, Bm,
        /*c_mod=*/(short)0, Cz, /*reuse_a=*/false, /*reuse_b=*/false);
    // lanes 0-15 see rows 0..7 in D[0..7]; lanes 16-31 see rows 8..15.
    float s = D[0] + D[1] + D[2] + D[3] + D[4] + D[5] + D[6] + D[7];
    s += __shfl_xor(s, 16, 32);   // combine the two half-wave partials

    __shared__ float wsum[8];     // 256 threads = 8 wave32
    const int lane = threadIdx.x & 31;
    const int wave = threadIdx.x >> 5;
    if (lane == 0) wsum[wave] = s;
    __syncthreads();
    if (threadIdx.x == 0) {
        float b = 0.0f;
#pragma unroll
        for (int i = 0; i < 8; ++i) b += wsum[i];
        block_sums[blockIdx.x] = b;
    }
}

__global__ void __launch_bounds__(256)
yolo_loss_final(const float* __restrict__ block_sums, int nb,
                float* __restrict__ out) {
    __shared__ double sm[256];
    double a = 0.0;
    for (int i = threadIdx.x; i < nb; i += blockDim.x) a += (double)block_sums[i];
    sm[threadIdx.x] = a;
    __syncthreads();
    for (int off = 128; off > 0; off >>= 1) {
        if (threadIdx.x < (unsigned)off) sm[threadIdx.x] += sm[threadIdx.x + off];
        __syncthreads();
    }
    if (threadIdx.x == 0) out[0] = (float)sm[0];
}

extern "C" void kernel_launch(void* const* d_in, const int* in_sizes, int n_in,
                              void* d_out, int out_size, void* d_ws, size_t ws_size,
                              hipStream_t stream) {
    const float* pred = (const float*)d_in[0];
    const float* targ = (const float*)d_in[1];
    float* out = (float*)d_out;
    float* block_sums = (float*)d_ws;

    const int n_cells = in_sizes[0] / 30;  // N*S*S
    const int NB = 1024, NT = 256;

    yolo_loss_main<<<NB, NT, 0, stream>>>(pred, targ, block_sums, n_cells);
    yolo_loss_final<<<1, 256, 0, stream>>>(block_sums, NB, out);
}